// MultiViewRecommender_88313117540623
// MI455X (gfx1250) — compile-verified
//
#include <hip/hip_runtime.h>
#include <math.h>

typedef __attribute__((ext_vector_type(2))) float v2f;
typedef __attribute__((ext_vector_type(8))) float v8f;

#define DIM 64
#define NCE_EPS 1e-8f

// ---------------- helpers ----------------

__device__ __forceinline__ float4 ld4(const float* p) { return *(const float4*)p; }

__device__ __forceinline__ float4 avg4(float4 a, float4 b) {
  float4 r;
  r.x = 0.5f * (a.x + b.x);
  r.y = 0.5f * (a.y + b.y);
  r.z = 0.5f * (a.z + b.z);
  r.w = 0.5f * (a.w + b.w);
  return r;
}

// Per-lane operand form for V_WMMA_F32_16X16X4_F32.
// A (16x4, lane=M) and B (4x16, lane=N) share the same per-lane layout:
//   lanes 0-15: VGPR0=K0, VGPR1=K1 ; lanes 16-31: VGPR0=K2, VGPR1=K3
__device__ __forceinline__ v2f pick(float4 q, bool hi) {
  v2f r;
  r.x = hi ? q.z : q.x;
  r.y = hi ? q.w : q.y;
  return r;
}

// In-lane |q|^2 accumulation: every lane sees its item's full row across the
// 16 chunks, so norms need no cross-lane reduction at all.
__device__ __forceinline__ float nrm4(float4 q, float acc) {
  return fmaf(q.x, q.x, fmaf(q.y, q.y, fmaf(q.z, q.z, fmaf(q.w, q.w, acc))));
}

// C layout: VGPR r: lanes 0-15 -> (M=r, N=lane); lanes 16-31 -> (M=r+8, N=lane-16).
// Diagonal element C[i][i] for item i = lane&15 sits IN-LANE at component lane&7,
// valid on lanes 0-7 (items 0-7) and 24-31 (items 8-15). Pure-VALU select chain.
__device__ __forceinline__ float diagsel(v8f c, int l8) {
  float d = c[0];
#pragma unroll
  for (int r = 1; r < 8; ++r) d = (l8 == r) ? c[r] : d;
  return d;
}

#define WMMA_F32(Aop, Bop, Cop) \
  __builtin_amdgcn_wmma_f32_16x16x4_f32(false, (Aop), false, (Bop), (short)0, (Cop), false, false)

// One InfoNCE view, single pass: softplus(cos(a1,an) - cos(a1,a2)) per item.
// WMMA computes the two cross dot products (diag of A x B^T); norms are plain
// in-lane FMA accumulation. Only 2 v8f accumulators live -> no spills.
template <bool AVG>
__device__ __forceinline__ float view_softplus(
    const float* a1p, const float* a1q,
    const float* a2p, const float* a2q,
    const float* anp, const float* anq,
    bool hi, int l8) {
  v8f c12 = {}, c1n = {};
  float n1 = 0.f, n2 = 0.f, nn = 0.f;
#pragma unroll 4
  for (int ch = 0; ch < 16; ++ch) {
    const int off = ch * 4;
    float4 q1 = ld4(a1p + off);
    float4 q2 = ld4(a2p + off);
    float4 qn = ld4(anp + off);
    if constexpr (AVG) {
      q1 = avg4(q1, ld4(a1q + off));
      q2 = avg4(q2, ld4(a2q + off));
      qn = avg4(qn, ld4(anq + off));
    }
    v2f f1 = pick(q1, hi);
    v2f f2 = pick(q2, hi);
    v2f fn = pick(qn, hi);
    c12 = WMMA_F32(f1, f2, c12);  // diag -> <a1_i, a2_i>
    c1n = WMMA_F32(f1, fn, c1n);  // diag -> <a1_i, an_i>
    n1 = nrm4(q1, n1);
    n2 = nrm4(q2, n2);
    nn = nrm4(qn, nn);
  }
  const float d12 = diagsel(c12, l8);
  const float d1n = diagsel(c1n, l8);
  const float rn1 = sqrtf(n1);
  const float pos = d12 / fmaxf(rn1 * sqrtf(n2), NCE_EPS);
  const float neg = d1n / fmaxf(rn1 * sqrtf(nn), NCE_EPS);
  const float x = neg - pos;
  return fmaxf(x, 0.f) + log1pf(expf(-fabsf(x)));  // stable softplus
}

// ---------------- kernels ----------------

// One wave32 handles 16 active items; lanes l and l+16 cooperate on item (lane&15).
__global__ void info_nce_kernel(const float* __restrict__ g1, const float* __restrict__ g2,
                                const float* __restrict__ s1, const float* __restrict__ s2,
                                const int* __restrict__ act, const int* __restrict__ ngi,
                                float* __restrict__ wsum, int A) {
  const int lane = (int)(threadIdx.x & 31);
  const int wid = (int)((blockIdx.x * blockDim.x + threadIdx.x) >> 5);
  if (wid * 16 >= A) return;  // wave-uniform exit
  const int l16 = lane & 15;  // item served by this lane (load/WMMA duty)
  const int l8 = lane & 7;    // diagonal component index
  const bool hi = lane >= 16;
  const bool isdiag = (lane < 8) || (lane >= 24);  // lanes holding diag elements
  const int k = wid * 16 + l16;
  const bool valid = isdiag && (k < A);
  const int kc = (k < A) ? k : (A - 1);  // clamp tail: load real rows, mask loss
  const size_t i = (size_t)act[kc] * DIM;
  const size_t j = (size_t)ngi[kc] * DIM;

  // graph view: a1=g1[i], a2=g2[i], an=g2[j]
  float sp = view_softplus<false>(g1 + i, nullptr, g2 + i, nullptr, g2 + j, nullptr, hi, l8);
  // seq view
  sp += view_softplus<false>(s1 + i, nullptr, s2 + i, nullptr, s2 + j, nullptr, hi, l8);
  // cross view: a1=g_avg[i], a2=s_avg[i], an=s_avg[j]
  sp += view_softplus<true>(g1 + i, g2 + i, s1 + i, s2 + i, s1 + j, s2 + j, hi, l8);

  float contrib = valid ? sp : 0.f;
#pragma unroll
  for (int off = 16; off; off >>= 1) contrib += __shfl_xor(contrib, off, 32);
  if (lane == 0) wsum[wid] = contrib;
}

// Deterministic fixed-order final reduction: L_CL = sum(wsum) / A.
__global__ void reduce_kernel(const float* __restrict__ wsum, int n,
                              float* __restrict__ loss, float scale) {
  __shared__ float sm[256];
  float s = 0.f;
  for (int idx = (int)threadIdx.x; idx < n; idx += 256) s += wsum[idx];
  sm[threadIdx.x] = s;
  __syncthreads();
  for (int off = 128; off; off >>= 1) {
    if ((int)threadIdx.x < off) sm[threadIdx.x] += sm[threadIdx.x + off];
    __syncthreads();
  }
  if (threadIdx.x == 0) *loss = sm[0] * scale;
}

// Wave-per-row gated fusion: out = alpha*g_avg + (1-alpha)*s_avg,
// alpha = sigmoid([g_avg,s_avg] @ att_w). Pure streaming, NT hints.
__global__ void fuse_kernel(const float* __restrict__ g1, const float* __restrict__ g2,
                            const float* __restrict__ s1, const float* __restrict__ s2,
                            const float* __restrict__ w, float* __restrict__ out, int N) {
  const int lane = (int)(threadIdx.x & 31);
  const int row = (int)((blockIdx.x * blockDim.x + threadIdx.x) >> 5);
  if (row >= N) return;  // wave-uniform
  const size_t base = (size_t)row * DIM + (size_t)(lane * 2);
  v2f a = __builtin_nontemporal_load((const v2f*)(g1 + base));
  v2f b = __builtin_nontemporal_load((const v2f*)(g2 + base));
  v2f c = __builtin_nontemporal_load((const v2f*)(s1 + base));
  v2f d = __builtin_nontemporal_load((const v2f*)(s2 + base));
  v2f ga, sa;
  ga.x = 0.5f * (a.x + b.x);
  ga.y = 0.5f * (a.y + b.y);
  sa.x = 0.5f * (c.x + d.x);
  sa.y = 0.5f * (c.y + d.y);
  const v2f wg = *(const v2f*)(w + 2 * lane);
  const v2f ws = *(const v2f*)(w + DIM + 2 * lane);
  float part = ga.x * wg.x + ga.y * wg.y + sa.x * ws.x + sa.y * ws.y;
#pragma unroll
  for (int off = 16; off; off >>= 1) part += __shfl_xor(part, off, 32);
  const float alpha = 1.f / (1.f + expf(-part));
  v2f o;
  o.x = alpha * ga.x + (1.f - alpha) * sa.x;
  o.y = alpha * ga.y + (1.f - alpha) * sa.y;
  __builtin_nontemporal_store(o, (v2f*)(out + base));
}

// ---------------- launch ----------------

extern "C" void kernel_launch(void* const* d_in, const int* in_sizes, int n_in,
                              void* d_out, int out_size, void* d_ws, size_t ws_size,
                              hipStream_t stream) {
  (void)n_in; (void)out_size; (void)ws_size;
  const float* g1 = (const float*)d_in[0];
  const float* g2 = (const float*)d_in[1];
  const float* s1 = (const float*)d_in[2];
  const float* s2 = (const float*)d_in[3];
  const float* w  = (const float*)d_in[4];
  const int* act  = (const int*)d_in[5];
  const int* ngi  = (const int*)d_in[6];
  const int N = in_sizes[0] / DIM;   // 500000
  const int A = in_sizes[5];         // 200000

  float* out  = (float*)d_out;
  float* loss = out + (size_t)N * DIM;   // L_CL stored after the [N,D] embeddings
  float* wsum = (float*)d_ws;            // one partial per 16-item wave (~50 KB)

  const int nWaves = (A + 15) / 16;
  const int block = 256;  // 8 wave32s
  const int grid_nce = (nWaves * 32 + block - 1) / block;
  info_nce_kernel<<<grid_nce, block, 0, stream>>>(g1, g2, s1, s2, act, ngi, wsum, A);
  reduce_kernel<<<1, 256, 0, stream>>>(wsum, nWaves, loss, 1.0f / (float)A);

  const int grid_fuse = (N + 7) / 8;  // 8 rows per block (wave-per-row)
  fuse_kernel<<<grid_fuse, block, 0, stream>>>(g1, g2, s1, s2, w, out, N);
}